// Change_length_57861799411997
// MI455X (gfx1250) — compile-verified
//
#include <hip/hip_runtime.h>
#include <math.h>

// Problem constants (fixed by setup_inputs in the reference).
#define B_   8
#define L_   4096
#define D_   256
#define T_   1024
#define KPAD 68    // max band rows per 16-j tile: ceil(16*4096/1024)+1 = 65 -> pad to 68
#define SSTR 272   // sSeq row stride (floats); 272 % 64 == 16 -> the two K-rows read by
                   // the two half-waves land in disjoint 16-bank groups (conflict-free)

typedef float v2f __attribute__((ext_vector_type(2)));
typedef float v8f __attribute__((ext_vector_type(8)));

// ---------------------------------------------------------------------------
// Kernel 1: per-batch stable compaction of the mask.
//   Lb[b]        = number of valid tokens
//   idxmap[b][p] = source row of p-th valid token (stable order)
// mask is a numpy bool array -> 1 byte per element.
// ---------------------------------------------------------------------------
__global__ void cl_compact_kernel(const unsigned char* __restrict__ mask,
                                  int* __restrict__ Lb,
                                  int* __restrict__ idxmap) {
  __shared__ int counts[256];
  const int b = blockIdx.x;
  const int t = threadIdx.x;
  const unsigned char* m = mask + (size_t)b * L_;
  const int base_l = t * 16;
  int c = 0;
  for (int i = 0; i < 16; ++i) c += (m[base_l + i] != 0);
  counts[t] = c;
  __syncthreads();
  if (t == 0) {
    int run = 0;
    for (int i = 0; i < 256; ++i) { int v = counts[i]; counts[i] = run; run += v; }
    Lb[b] = run;
  }
  __syncthreads();
  int pos = counts[t];
  int* im = idxmap + (size_t)b * L_;
  for (int i = 0; i < 16; ++i) {
    int l = base_l + i;
    if (m[l] != 0) im[pos++] = l;
  }
}

// ---------------------------------------------------------------------------
// Kernel 2: banded pooling via fp32 WMMA with async-to-LDS staging.
//   Block = (jt, b): 16 output rows, 256 channels, 128 threads = 4 waves.
//   Wave w owns channel tiles d = [64w, 64w+64) as four 16-wide WMMA tiles.
// ---------------------------------------------------------------------------
__global__ __launch_bounds__(128) void cl_pool_wmma_kernel(
    const float* __restrict__ x,
    const int* __restrict__ Lb,
    const int* __restrict__ idxmap,
    float* __restrict__ out) {
  extern __shared__ float smem[];
  float* sSeq = smem;                    // [KPAD][SSTR] gathered rows of x
  float* sWA  = smem + KPAD * SSTR;      // [16][KPAD]   pool weights (1/bin)
  float* sWW  = sWA + 16 * KPAD;         // [16][KPAD]   overlap weights (normalized)
  float* sInv = sWW + 16 * KPAD;         // [16]

  const int jt  = blockIdx.x;
  const int b   = blockIdx.y;
  const int j0  = jt * 16;
  const int tid = threadIdx.x;
  const int lb  = Lb[b];

  // Band start. j*lb <= 2^22 and T=1024 is a power of two, so this integer
  // floor matches the reference's fp32 floor exactly.
  const int kstart = (j0 * lb) >> 10;
  int kvalid = lb - kstart; if (kvalid > KPAD) kvalid = KPAD;

  // ---- 1) issue async DMA of the band into LDS (overlaps weight math) ----
  {
    const int*   im = idxmap + (size_t)b * L_;
    const float* xb = x + (size_t)b * L_ * D_;
    const unsigned lds_base = (unsigned)(uintptr_t)sSeq;  // low 32b = LDS offset
    for (int kk0 = 0; kk0 < KPAD; kk0 += 2) {             // 2 rows per pass
      const int kk = kk0 + (tid >> 6);
      const int c  = (tid & 63) * 4;                      // float4 chunk
      const unsigned lds_off = lds_base + (unsigned)((kk * SSTR + c) * 4);
      if (kk < kvalid) {
        const unsigned long long gaddr =
            (unsigned long long)(uintptr_t)(xb + (size_t)im[kstart + kk] * D_ + c);
        asm volatile("global_load_async_to_lds_b128 %0, %1, off"
                     :: "v"(lds_off), "v"(gaddr) : "memory");
      } else {
        *(float4*)(sSeq + kk * SSTR + c) = make_float4(0.f, 0.f, 0.f, 0.f);
      }
    }
  }

  // ---- 2) build weight matrices (16 x KPAD) while the DMA runs ----
  const float step = (float)lb * (1.0f / 1024.0f);        // exact in fp32
  for (int e = tid; e < 16 * KPAD; e += 128) {
    const int jj = e / KPAD;
    const int kk = e - jj * KPAD;
    const int j  = j0 + jj;
    const int l  = kstart + kk;
    const int sA = (j * lb) >> 10;                        // floor(j*lb/T)
    const int eA = ((j + 1) * lb + 1023) >> 10;           // ceil((j+1)*lb/T)
    int den = eA - sA; if (den < 1) den = 1;
    sWA[e] = (l >= sA && l < eA) ? (1.0f / (float)den) : 0.0f;
    const float st = (float)j * step;
    const float en = st + step;
    const float fl = (float)l;
    float w = fminf(fl + 1.0f, en) - fmaxf(fl, st);
    sWW[e] = fmaxf(w, 0.0f);
  }
  __syncthreads();
  if (tid < 16) {
    float s = 0.0f;
    for (int kk = 0; kk < KPAD; ++kk) s += sWW[tid * KPAD + kk];
    sInv[tid] = 1.0f / fmaxf(s, 1e-12f);
  }
  __syncthreads();
  for (int e = tid; e < 16 * KPAD; e += 128) sWW[e] *= sInv[e / KPAD];
  // DMA must be complete and weights visible before the MMA loop.
  asm volatile("s_wait_asynccnt 0x0" ::: "memory");
  __syncthreads();

  // ---- 3) WMMA loop: K outer, 4 channel tiles unrolled (A reused 12x) ----
  const int lane = tid & 31;
  const int wave = tid >> 5;
  const int half = lane >> 4;     // K-pair select: (0,1) vs (2,3)
  const int idx  = lane & 15;     // M for A, N for B
  const int d0   = wave * 64;

  v8f accP[4], accM[4], accQ[4];
#pragma unroll
  for (int t = 0; t < 4; ++t) { accP[t] = (v8f){}; accM[t] = (v8f){}; accQ[t] = (v8f){}; }

  for (int kb = 0; kb < KPAD; kb += 4) {
    const int k0 = kb + 2 * half;                 // even -> 8B-aligned b64 loads
    const v2f aA = *(const v2f*)(sWA + idx * KPAD + k0);
    const v2f aW = *(const v2f*)(sWW + idx * KPAD + k0);
    const float* r0 = sSeq + k0 * SSTR + d0 + idx;
    const float* r1 = r0 + SSTR;
#pragma unroll
    for (int t = 0; t < 4; ++t) {
      v2f bv, bq;
      bv.x = r0[t * 16];
      bv.y = r1[t * 16];
      bq.x = bv.x * bv.x;
      bq.y = bv.y * bv.y;
      accP[t] = __builtin_amdgcn_wmma_f32_16x16x4_f32(false, aA, false, bv,
                                                      (short)0, accP[t], false, false);
      accM[t] = __builtin_amdgcn_wmma_f32_16x16x4_f32(false, aW, false, bv,
                                                      (short)0, accM[t], false, false);
      accQ[t] = __builtin_amdgcn_wmma_f32_16x16x4_f32(false, aW, false, bq,
                                                      (short)0, accQ[t], false, false);
    }
  }

  // ---- 4) write padded_out, std_out, out_mask ----
  float* padded = out;                          // [B,T,D]
  float* omask  = out + (size_t)B_ * T_ * D_;   // [B,T] (bool -> 1.0f)
  float* stdo   = omask + (size_t)B_ * T_;      // [B,T,D]
#pragma unroll
  for (int t = 0; t < 4; ++t) {
    const int d = d0 + t * 16 + idx;
#pragma unroll
    for (int r = 0; r < 8; ++r) {
      const int jg = j0 + r + 8 * half;         // C layout: VGPR r -> M = r + 8*half
      const size_t o = ((size_t)(b * T_ + jg) << 8) + d;
      padded[o] = accP[t][r];
      const float mv = accM[t][r];
      float var = accQ[t][r] - mv * mv;
      var = fmaxf(var, 1e-12f);
      stdo[o] = sqrtf(var);
    }
  }
  if (tid < 16) omask[b * T_ + j0 + tid] = 1.0f;
}

// ---------------------------------------------------------------------------
extern "C" void kernel_launch(void* const* d_in, const int* in_sizes, int n_in,
                              void* d_out, int out_size, void* d_ws, size_t ws_size,
                              hipStream_t stream) {
  const float*         x    = (const float*)d_in[0];
  const unsigned char* mask = (const unsigned char*)d_in[1];  // numpy bool: 1B/elem
  // d_in[2] = finallength (fixed 1024 -> compile-time T_)

  int* LbP    = (int*)d_ws;
  int* idxmap = (int*)((char*)d_ws + 64);

  cl_compact_kernel<<<dim3(B_), dim3(256), 0, stream>>>(mask, LbP, idxmap);

  const size_t shmem = (size_t)(KPAD * SSTR + 2 * 16 * KPAD + 16) * sizeof(float);
  dim3 grid(T_ / 16, B_);
  cl_pool_wmma_kernel<<<grid, dim3(128), shmem, stream>>>(x, LbP, idxmap, (float*)d_out);
}